// GenericEncoder_22084721836480
// MI455X (gfx1250) — compile-verified
//
#include <hip/hip_runtime.h>

typedef __attribute__((ext_vector_type(2))) float v2f;
typedef __attribute__((ext_vector_type(8))) float v8f;

// ---------------------------------------------------------------------------
// Small elementwise kernels
// ---------------------------------------------------------------------------
__global__ void zero_f32(float* __restrict__ p, int n) {
    int i = blockIdx.x * blockDim.x + threadIdx.x;
    if (i < n) p[i] = 0.0f;
}

__global__ void deg_count(const int* __restrict__ dst, float* __restrict__ deg, int E) {
    int e = blockIdx.x * blockDim.x + threadIdx.x;
    if (e < E) atomicAdd(&deg[dst[e]], 1.0f);
}

// dis[i] = rsqrt(deg_raw[i] + 1)   (self-loop contributes +1; result always in (0, 1])
__global__ void deg_to_dis(float* __restrict__ deg, int n) {
    int i = blockIdx.x * blockDim.x + threadIdx.x;
    if (i < n) deg[i] = rsqrtf(deg[i] + 1.0f);
}

// p[i] = bias[i & mask]  (C is a power of two)
__global__ void fill_bias(float* __restrict__ p, const float* __restrict__ bias,
                          int mask, long long total) {
    long long i = (long long)blockIdx.x * blockDim.x + threadIdx.x;
    if (i < total) p[i] = bias[(int)(i & (long long)mask)];
}

__global__ void relu_inplace(float* __restrict__ p, long long total) {
    long long i = (long long)blockIdx.x * blockDim.x + threadIdx.x;
    if (i < total) p[i] = fmaxf(p[i], 0.0f);
}

// ---------------------------------------------------------------------------
// fp32 WMMA GEMM:  C[N x M] = A[N x K] @ B[K x M]
// One wave per 16x16 output tile; K consumed 4 at a time with
// v_wmma_f32_16x16x4_f32 (exact fp32, matches the fp32 reference).
// A-frag (ISA 7.12.2, 32-bit 16x4): lanes 0-15 -> rows, half selects K pair:
//   v0 = K = k0+2*half, v1 = K = k0+2*half+1
// B-frag mirrored (row of B striped across lanes within a VGPR).
// D tile: VGPR v -> row (8*half + v), col = lane&15.
// ---------------------------------------------------------------------------
__global__ void __launch_bounds__(128)
wmma_gemm_f32(const float* __restrict__ A, const float* __restrict__ B,
              float* __restrict__ C, int N, int K, int M) {
    const int lane   = threadIdx.x & 31;
    const int mTiles = M >> 4;
    const int tile   = blockIdx.x * blockDim.y + threadIdx.y;
    const int rowT   = tile / mTiles;
    const int colT   = tile - rowT * mTiles;
    if (rowT >= (N >> 4)) return;               // uniform per wave

    const int row0 = rowT << 4;
    const int col0 = colT << 4;
    const int half = lane >> 4;                  // 0: K pair {0,1}, 1: {2,3}
    const int lid  = lane & 15;

    const float* Arow = A + (long long)(row0 + lid) * K;
    const float* Bcol = B + col0 + lid;

    v8f acc = {};
    #pragma unroll 4
    for (int k0 = 0; k0 < K; k0 += 4) {
        const int ka = k0 + (half << 1);
        v2f a, b;
        a.x = Arow[ka];
        a.y = Arow[ka + 1];
        b.x = Bcol[(long long)ka * M];
        b.y = Bcol[(long long)(ka + 1) * M];
        acc = __builtin_amdgcn_wmma_f32_16x16x4_f32(
            /*neg_a=*/false, a, /*neg_b=*/false, b,
            /*c_mod=*/(short)0, acc, /*reuse_a=*/false, /*reuse_b=*/false);
    }

    float* Crow = C + (long long)(row0 + (half << 3)) * M + col0 + lid;
    #pragma unroll
    for (int v = 0; v < 8; ++v)
        Crow[(long long)v * M] = acc[v];
}

// ---------------------------------------------------------------------------
// Edge scatter:  out[dst] += T[src] * (dis[src]*dis[dst])   over E real edges
// plus N self-loops (norm = dis[i]^2).  One wave per edge, lane = channel
// (stride 32) so gathers and atomics are fully coalesced; tables are
// L2-resident (192 MB L2 >> 51 MB hW).
// ---------------------------------------------------------------------------
template <int C>
__global__ void __launch_bounds__(256)
edge_scatter(const float* __restrict__ T, const int* __restrict__ src,
             const int* __restrict__ dst, const float* __restrict__ dis,
             float* __restrict__ out, int E, int N) {
    const int e    = blockIdx.x * (blockDim.x >> 5) + (threadIdx.x >> 5);
    const int lane = threadIdx.x & 31;
    if (e >= E + N) return;

    int s, d;
    float nrm;
    if (e < E) {
        s = src[e];
        d = dst[e];
        nrm = dis[s] * dis[d];
    } else {                 // self loop
        s = d = e - E;
        const float di = dis[s];
        nrm = di * di;
    }
    const float* row  = T + (long long)s * C;
    float*       orow = out + (long long)d * C;
    #pragma unroll
    for (int i = 0; i < C / 32; ++i) {
        const int c = lane + 32 * i;
        atomicAdd(&orow[c], row[c] * nrm);
    }
}

// ---------------------------------------------------------------------------
// Host-side launch sequence
// ---------------------------------------------------------------------------
static inline int ceil_div_ll(long long a, int b) { return (int)((a + b - 1) / b); }

extern "C" void kernel_launch(void* const* d_in, const int* in_sizes, int n_in,
                              void* d_out, int out_size, void* d_ws, size_t ws_size,
                              hipStream_t stream) {
    const float* x   = (const float*)d_in[0];
    const int*   ei  = (const int*)  d_in[1];
    const float* W1  = (const float*)d_in[2];
    const float* b1  = (const float*)d_in[3];
    const float* Wmu = (const float*)d_in[4];
    const float* bmu = (const float*)d_in[5];
    const float* Wlv = (const float*)d_in[6];
    const float* blv = (const float*)d_in[7];

    const int CIN = 128, CHID = 128, COUT = 64;
    const int N = in_sizes[0] / CIN;     // 100000
    const int E = in_sizes[1] / 2;       // 1600000
    const int* src = ei;
    const int* dst = ei + E;

    float* mu = (float*)d_out;                       // [N, 64]
    float* lv = mu + (long long)N * COUT;            // [N, 64]

    // workspace: dis[N] | hW[N*128] (reused as t[N*64]) | h[N*128]
    float* dis = (float*)d_ws;
    float* hW  = dis + N;
    float* h   = hW + (long long)N * CHID;

    const long long nh  = (long long)N * CHID;       // 12.8M
    const long long no  = (long long)N * COUT;       // 6.4M
    const int EN_blocks = ceil_div_ll((long long)E + N, 8);   // 8 edges/block

    // 1) degree + symmetric normalization
    zero_f32<<<ceil_div_ll(N, 256), 256, 0, stream>>>(dis, N);
    deg_count<<<ceil_div_ll(E, 256), 256, 0, stream>>>(dst, dis, E);
    deg_to_dis<<<ceil_div_ll(N, 256), 256, 0, stream>>>(dis, N);

    // 2) hW = x @ W1   (fp32 WMMA)
    {
        dim3 blk(32, 4);
        int tiles = (N >> 4) * (CHID >> 4);          // 6250*8
        wmma_gemm_f32<<<ceil_div_ll(tiles, 4), blk, 0, stream>>>(x, W1, hW, N, CIN, CHID);
    }
    // 3) h = relu(scatter(hW) + b1)
    fill_bias<<<ceil_div_ll(nh, 256), 256, 0, stream>>>(h, b1, CHID - 1, nh);
    edge_scatter<128><<<EN_blocks, 256, 0, stream>>>(hW, src, dst, dis, h, E, N);
    relu_inplace<<<ceil_div_ll(nh, 256), 256, 0, stream>>>(h, nh);

    // 4) mu = scatter(h @ Wmu) + bmu
    {
        dim3 blk(32, 4);
        int tiles = (N >> 4) * (COUT >> 4);          // 6250*4
        wmma_gemm_f32<<<ceil_div_ll(tiles, 4), blk, 0, stream>>>(h, Wmu, hW, N, CHID, COUT);
        fill_bias<<<ceil_div_ll(no, 256), 256, 0, stream>>>(mu, bmu, COUT - 1, no);
        edge_scatter<64><<<EN_blocks, 256, 0, stream>>>(hW, src, dst, dis, mu, E, N);
    }
    // 5) logvar = scatter(h @ Wlv) + blv
    {
        dim3 blk(32, 4);
        int tiles = (N >> 4) * (COUT >> 4);
        wmma_gemm_f32<<<ceil_div_ll(tiles, 4), blk, 0, stream>>>(h, Wlv, hW, N, CHID, COUT);
        fill_bias<<<ceil_div_ll(no, 256), 256, 0, stream>>>(lv, blv, COUT - 1, no);
        edge_scatter<64><<<EN_blocks, 256, 0, stream>>>(hW, src, dst, dis, lv, E, N);
    }
}